// CoupledStateSpaceVI_7808250544202
// MI455X (gfx1250) — compile-verified
//
#include <hip/hip_runtime.h>
#include <hip/hip_bf16.h>
#include <math.h>

#define TT      131072
#define DTC     (1.0f/1024.0f)
#define LOG2PI_F 1.8378770664093453f

typedef __attribute__((ext_vector_type(2))) float v2f;
typedef __attribute__((ext_vector_type(8))) float v8f;

__device__ __forceinline__ v8f wmma4(v2f a, v2f b, v8f c) {
  // D = A(16x4 f32) x B(4x16 f32) + C(16x16 f32)
  return __builtin_amdgcn_wmma_f32_16x16x4_f32(false, a, false, b, (short)0, c, false, false);
}

__device__ __forceinline__ float block_reduce256(float v, float* red, int tid) {
  red[tid] = v;
  __syncthreads();
  for (int s = 128; s > 0; s >>= 1) {
    if (tid < s) red[tid] += red[tid + s];
    __syncthreads();
  }
  float r = red[0];
  __syncthreads();
  return r;
}

// ---------------------------------------------------------------------------
// Kernel 1: R^-1 (Gauss-Jordan), K = R^-1 B^T, A_aug, z_b, and the three
// scalar loss terms -> elbo. Single 256-thread block.
// ---------------------------------------------------------------------------
__global__ void setup_kernel(const float* __restrict__ y, const float* __restrict__ C,
                             const float* __restrict__ A, const float* __restrict__ B,
                             const float* __restrict__ Q, const float* __restrict__ R,
                             const float* __restrict__ q_mu, const float* __restrict__ q_logvar,
                             const float* __restrict__ eps_batch,
                             float* __restrict__ Aaug, float* __restrict__ Kmat,
                             float* __restrict__ zb, float* __restrict__ elbo_out) {
  __shared__ float M[64][130];   // [R | I] augmented, padded
  __shared__ float Ksh[64][65];
  __shared__ float red[256];
  const int tid = threadIdx.x;

  for (int idx = tid; idx < 64 * 128; idx += 256) {
    int i = idx >> 7, j = idx & 127;
    M[i][j] = (j < 64) ? R[i * 64 + j] : ((j - 64) == i ? 1.0f : 0.0f);
  }
  __syncthreads();

  // Gauss-Jordan (no pivot search; R is well-conditioned / identity here)
  for (int p = 0; p < 64; ++p) {
    float inv = 1.0f / M[p][p];
    __syncthreads();
    if (tid < 128) M[p][tid] *= inv;
    __syncthreads();
    if (tid < 64 && tid != p) {
      float f = M[tid][p];
      for (int j = 0; j < 128; ++j) M[tid][j] = fmaf(-f, M[p][j], M[tid][j]);
    }
    __syncthreads();
  }

  // K = R^-1 @ B^T
  for (int idx = tid; idx < 64 * 64; idx += 256) {
    int i = idx >> 6, j = idx & 63;
    float s = 0.0f;
    for (int k = 0; k < 64; ++k) s = fmaf(M[i][64 + k], B[j * 64 + k], s);
    Ksh[i][j] = s;
    Kmat[i * 64 + j] = s;
  }
  __syncthreads();

  // A_aug = [[A, -B R^-1 B^T], [-Q, -A^T]]  (BRB = B @ K)
  for (int idx = tid; idx < 128 * 128; idx += 256) {
    int i = idx >> 7, j = idx & 127;
    float v;
    if (i < 64) {
      if (j < 64) {
        v = A[i * 64 + j];
      } else {
        int jj = j - 64;
        float s = 0.0f;
        for (int k = 0; k < 64; ++k) s = fmaf(B[i * 64 + k], Ksh[k][jj], s);
        v = -s;
      }
    } else {
      int ii = i - 64;
      v = (j < 64) ? -Q[ii * 64 + j] : -A[(j - 64) * 64 + ii];
    }
    Aaug[i * 128 + j] = v;
  }

  // z_b and KL
  float klsum = 0.0f;
  for (int idx = tid; idx < 128 * 128; idx += 256) {
    float mu = q_mu[idx], lv = q_logvar[idx];
    klsum += 1.0f + lv - (mu * mu + expf(lv));
    zb[idx] = fmaf(eps_batch[idx], expf(0.5f * lv), mu);
  }
  __threadfence();
  __syncthreads();

  // recon: sum((y[:128] - x_b C^T)^2)
  float sse = 0.0f;
  for (int idx = tid; idx < 128 * 64; idx += 256) {
    int t = idx >> 6, j = idx & 63;
    float s = 0.0f;
    for (int k = 0; k < 64; ++k) s = fmaf(zb[t * 128 + k], C[j * 64 + k], s);
    float e = y[idx] - s;
    sse = fmaf(e, e, sse);
  }

  // dynamics: one thread per time-row
  float dyn = 0.0f;
  for (int t = tid; t < 127; t += 256) {
    float ssq = 0.0f;
    for (int c = 0; c < 128; ++c) {
      float dot = 0.0f;
      for (int k = 0; k < 128; ++k) dot = fmaf(zb[t * 128 + k], Aaug[c * 128 + k], dot);
      float pred = fmaf(DTC, dot, zb[t * 128 + c]);
      float de = zb[(t + 1) * 128 + c] - pred;
      ssq = fmaf(de, de, ssq);
    }
    float term = -0.5f * (ssq + 128.0f * LOG2PI_F);
    if (__builtin_isfinite(term)) dyn += term;
  }

  float klT  = block_reduce256(klsum, red, tid);
  float sseT = block_reduce256(sse, red, tid);
  float dynT = block_reduce256(dyn, red, tid);
  if (tid == 0) {
    float kl = -0.5f * klT;
    float recon = -0.5f * (sseT + 8192.0f * LOG2PI_F);
    elbo_out[0] = recon + dynT - 0.1f * kl;
  }
}

// ---------------------------------------------------------------------------
// Kernel 2: sequential Riccati-like scan, 127 steps, all matrices in LDS,
// WMMA f32 16x16x4 for X_new = RHS + DT*(A^T X + X A). One WG, 32 waves.
// ---------------------------------------------------------------------------
__global__ void riccati_kernel(const float* __restrict__ Aaug, const float* __restrict__ Q,
                               const float* __restrict__ P0, float* __restrict__ Pout) {
  extern __shared__ float lds[];
  const int P = 130;                 // padded row pitch (floats)
  float* As = lds;                   // A_aug           128*130
  float* RH = lds + 128 * P;         // RHS             128*130
  float* Xc = lds + 2 * 128 * P;     // ping
  float* Xn = lds + 3 * 128 * P;     // pong
  const int tid = threadIdx.x;
  const int lane = tid & 31, half = lane >> 4, l16 = lane & 15;
  const int wave = tid >> 5;

  for (int idx = tid; idx < 128 * 128; idx += 1024) {
    int i = idx >> 7, j = idx & 127;
    As[i * P + j] = Aaug[idx];
    Xc[i * P + j] = P0[idx];
  }
  __syncthreads();

  for (int step = 0; step < 127; ++step) {
    // RHS = P + DT*Q_aug  (Q_aug nonzero only in [:64,:64])
    for (int idx = tid; idx < 128 * 128; idx += 1024) {
      int i = idx >> 7, j = idx & 127;
      float v = Xc[i * P + j];
      if (i < 64 && j < 64) v = fmaf(DTC, Q[i * 64 + j], v);
      RH[i * P + j] = v;
    }
    __syncthreads();

    for (int it = 0; it < 3; ++it) {
      // each wave owns 2 of the 64 output 16x16 tiles
      for (int t = wave * 2; t < wave * 2 + 2; ++t) {
        const int i0 = (t >> 3) * 16, j0 = (t & 7) * 16;
        v8f acc = {0.f, 0.f, 0.f, 0.f, 0.f, 0.f, 0.f, 0.f};
        for (int k0 = 0; k0 < 128; k0 += 4) {
          const int ka = k0 + 2 * half;
          v2f a1, b1, a2, b2;
          // A^T X : Afrag[m][k] = A[k][i0+m], Bfrag[k][n] = X[k][j0+n]
          a1.x = As[ka * P + i0 + l16];       a1.y = As[(ka + 1) * P + i0 + l16];
          b1.x = Xc[ka * P + j0 + l16];       b1.y = Xc[(ka + 1) * P + j0 + l16];
          acc = wmma4(a1, b1, acc);
          // X A : Afrag[m][k] = X[i0+m][k], Bfrag[k][n] = A[k][j0+n]
          a2.x = Xc[(i0 + l16) * P + ka];     a2.y = Xc[(i0 + l16) * P + ka + 1];
          b2.x = As[ka * P + j0 + l16];       b2.y = As[(ka + 1) * P + j0 + l16];
          acc = wmma4(a2, b2, acc);
        }
        for (int r = 0; r < 8; ++r) {
          const int gi = i0 + r + 8 * half, gj = j0 + l16;
          Xn[gi * P + gj] = fmaf(DTC, acc[r], RH[gi * P + gj]);
        }
      }
      __syncthreads();
      float* tmp = Xc; Xc = Xn; Xn = tmp;
    }

    // P_new = 0.5*(X + X^T), in place (each i<j pair owned by one thread)
    for (int idx = tid; idx < 128 * 128; idx += 1024) {
      int i = idx >> 7, j = idx & 127;
      if (i < j) {
        float a = Xc[i * P + j], b = Xc[j * P + i];
        float m = 0.5f * (a + b);
        Xc[i * P + j] = m;
        Xc[j * P + i] = m;
      }
    }
    __syncthreads();
  }

  for (int idx = tid; idx < 128 * 128; idx += 1024) {
    int i = idx >> 7, j = idx & 127;
    Pout[idx] = Xc[i * P + j];
  }
}

// ---------------------------------------------------------------------------
// Kernel 3: streaming part (HBM-bound). One block per 16 rows of T.
// b128 global loads -> z tile in LDS, emit x/p, then WMMA f32 16x16x4 for
// y_pred = x C^T and u = -(p K^T). Output stores stay b32 (4B-aligned bases).
// ---------------------------------------------------------------------------
__global__ void stream_kernel(const float* __restrict__ q_mu, const float* __restrict__ q_logvar,
                              const float* __restrict__ eps_full, const float* __restrict__ C,
                              const float* __restrict__ Kmat,
                              float* __restrict__ xo, float* __restrict__ po,
                              float* __restrict__ uo, float* __restrict__ yp) {
  __shared__ float zt[16 * 132];     // pitch 132: float4-aligned rows, bank stride 4
  __shared__ float Cs[64 * 68];
  __shared__ float Ks[64 * 68];
  const int tid = threadIdx.x;
  const int lane = tid & 31, half = lane >> 4, l16 = lane & 15;
  const int w = tid >> 5;            // 4 waves -> 4 column tiles
  const long row0 = (long)blockIdx.x * 16;

  const float4* C4   = (const float4*)C;
  const float4* K4   = (const float4*)Kmat;
  const float4* mu4  = (const float4*)q_mu;
  const float4* lv4  = (const float4*)q_logvar;
  const float4* ep4  = (const float4*)eps_full;

  // preload C and K (64x64 each) as float4
  for (int idx = tid; idx < 1024; idx += 128) {
    int i = idx >> 4, j4 = (idx & 15) << 2;
    *(float4*)&Cs[i * 68 + j4] = C4[idx];
    *(float4*)&Ks[i * 68 + j4] = K4[idx];
  }
  // z = mu + eps*exp(0.5*lv) for 16 rows x 128 cols, 4 floats per iteration
  for (int idx = tid; idx < 512; idx += 128) {
    int r = idx >> 5, c4 = idx & 31;             // c4: group of 4 columns
    long g4 = (row0 + r) * 32 + c4;
    float4 mu = mu4[g4], lv = lv4[g4], ep = ep4[g4];
    float4 z;
    z.x = fmaf(ep.x, expf(0.5f * lv.x), mu.x);
    z.y = fmaf(ep.y, expf(0.5f * lv.y), mu.y);
    z.z = fmaf(ep.z, expf(0.5f * lv.z), mu.z);
    z.w = fmaf(ep.w, expf(0.5f * lv.w), mu.w);
    *(float4*)&zt[r * 132 + (c4 << 2)] = z;
    // x = z[:, :64], p = z[:, 64:]  (output base only 4B-aligned -> b32 stores)
    long o = (row0 + r) * 64;
    int c = c4 << 2;
    float* dst = (c < 64) ? (xo + o + c) : (po + o + c - 64);
    dst[0] = z.x; dst[1] = z.y; dst[2] = z.z; dst[3] = z.w;
  }
  __syncthreads();

  v8f accY = {0.f, 0.f, 0.f, 0.f, 0.f, 0.f, 0.f, 0.f};
  v8f accU = {0.f, 0.f, 0.f, 0.f, 0.f, 0.f, 0.f, 0.f};
  const int n = w * 16 + l16;        // output column (0..63)
  for (int k0 = 0; k0 < 64; k0 += 4) {
    const int ka = k0 + 2 * half;
    v2f ax, bc, ap, bk;
    // y_pred[m][n] = sum_k x[m][k] * C[n][k]
    ax.x = zt[l16 * 132 + ka];       ax.y = zt[l16 * 132 + ka + 1];
    bc.x = Cs[n * 68 + ka];          bc.y = Cs[n * 68 + ka + 1];
    accY = wmma4(ax, bc, accY);
    // u[m][n] = -sum_k p[m][k] * K[n][k]
    ap.x = zt[l16 * 132 + 64 + ka];  ap.y = zt[l16 * 132 + 64 + ka + 1];
    bk.x = Ks[n * 68 + ka];          bk.y = Ks[n * 68 + ka + 1];
    accU = wmma4(ap, bk, accU);
  }
  for (int r = 0; r < 8; ++r) {
    long gi = row0 + r + 8 * half;
    yp[gi * 64 + n] = accY[r];
    uo[gi * 64 + n] = -accU[r];
  }
}

// ---------------------------------------------------------------------------
extern "C" void kernel_launch(void* const* d_in, const int* in_sizes, int n_in,
                              void* d_out, int out_size, void* d_ws, size_t ws_size,
                              hipStream_t stream) {
  (void)in_sizes; (void)n_in; (void)out_size; (void)ws_size;
  const float* y     = (const float*)d_in[0];
  const float* C     = (const float*)d_in[1];
  const float* A     = (const float*)d_in[2];
  const float* B     = (const float*)d_in[3];
  const float* Q     = (const float*)d_in[4];
  const float* R     = (const float*)d_in[5];
  const float* q_mu  = (const float*)d_in[6];
  const float* q_lv  = (const float*)d_in[7];
  const float* P0    = (const float*)d_in[8];
  const float* eps_b = (const float*)d_in[9];
  const float* eps_f = (const float*)d_in[10];

  float* out = (float*)d_out;
  float* ws  = (float*)d_ws;
  float* Aaug = ws;              // 16384
  float* Kmat = ws + 16384;      // 4096
  float* zb   = ws + 20480;      // 16384

  const long n = (long)TT * 64;
  float* elbo = out;
  float* Pout = out + 1;
  float* xo   = out + 1 + 16384;
  float* po   = xo + n;
  float* uo   = po + n;
  float* yp   = uo + n;

  setup_kernel<<<1, 256, 0, stream>>>(y, C, A, B, Q, R, q_mu, q_lv, eps_b,
                                      Aaug, Kmat, zb, elbo);
  const size_t shmem2 = 4ull * 128 * 130 * sizeof(float);  // 266240 B of 320KB LDS
  riccati_kernel<<<1, 1024, shmem2, stream>>>(Aaug, Q, P0, Pout);
  stream_kernel<<<TT / 16, 128, 0, stream>>>(q_mu, q_lv, eps_f, C, Kmat,
                                             xo, po, uo, yp);
}